// DualTransitionStepMixin_41317585387619
// MI455X (gfx1250) — compile-verified
//
#include <hip/hip_runtime.h>
#include <hip/hip_bf16.h>

typedef float v2f __attribute__((ext_vector_type(2)));
typedef float v8f __attribute__((ext_vector_type(8)));

#define BB 64
#define HH 96
#define DD 32
#define NT 192          // 6 wave32 waves per workgroup
#define CSTRIDE 33      // LDS padding: 33*row is invertible mod 64 banks

__launch_bounds__(NT)
__global__ void dual_transition_kernel(const float* __restrict__ lst,
                                       const float* __restrict__ init_logits,
                                       const float* __restrict__ trans_logits,
                                       float* __restrict__ out_est,
                                       float* __restrict__ out_ctrl) {
  __shared__ float sT[DD][CSTRIDE];   // row-stochastic T = softmax(trans_logits, -1)
  __shared__ float sC[HH][CSTRIDE];   // ctrl rows, evolves C <- C*T + (C e) q^T
  __shared__ float sE[DD], sR[DD], sQ[DD], sW[DD], sS[DD], sP[DD];
  __shared__ float sDots[HH];

  const int b    = blockIdx.x;
  const int t    = threadIdx.x;
  const int wave = t >> 5;
  const int lane = t & 31;
  const int lo   = lane & 15;
  const int hi   = lane >> 4;

  // ---------------- setup ----------------
  for (int i = t; i < HH * CSTRIDE; i += NT) (&sC[0][0])[i] = 0.0f;

  if (t < DD) {                 // softmax of transition row t
    float m = -3.4e38f;
    for (int j = 0; j < DD; ++j) m = fmaxf(m, trans_logits[t * DD + j]);
    float s = 0.0f;
    for (int j = 0; j < DD; ++j) s += __expf(trans_logits[t * DD + j] - m);
    float inv = 1.0f / s;
    for (int j = 0; j < DD; ++j) sT[t][j] = __expf(trans_logits[t * DD + j] - m) * inv;
  } else if (t == DD) {         // softmax of initial logits -> p0, w_{-1}, s_{-1}
    float m = -3.4e38f;
    for (int j = 0; j < DD; ++j) m = fmaxf(m, init_logits[j]);
    float s = 0.0f;
    for (int j = 0; j < DD; ++j) s += __expf(init_logits[j] - m);
    float inv = 1.0f / s;
    for (int j = 0; j < DD; ++j) {
      float v = __expf(init_logits[j] - m) * inv;
      sP[j] = v; sW[j] = v; sS[j] = 0.0f;
    }
  }
  __syncthreads();

  // Hoist the constant B-operand (T) fragments for all 8 K-chunks x 2 col-tiles.
  // B 4x16 layout: k = vgpr + 2*(lane/16), col = lane%16.
  v2f bfr[8][2];
#pragma unroll
  for (int kc = 0; kc < 8; ++kc) {
    const int k0 = kc * 4;
    bfr[kc][0].x = sT[k0 + 2 * hi][lo];
    bfr[kc][0].y = sT[k0 + 2 * hi + 1][lo];
    bfr[kc][1].x = sT[k0 + 2 * hi][lo + 16];
    bfr[kc][1].y = sT[k0 + 2 * hi + 1][lo + 16];
  }

  float wdot = 0.0f, sdot = 0.0f;   // live in wave0 registers across phases

  for (int k = 0; k < HH; ++k) {
    // ---- phase 0: e_k = 2*lst - 1 ----
    if (t < DD) sE[t] = 2.0f * lst[(b * HH + k) * DD + t] - 1.0f;
    __syncthreads();

    // ---- phase 1: row dots (C e), and wave0 reductions for ee / w.e / s.e, r ----
    if (t < HH) {
      float acc = 0.0f;
      for (int n = 0; n < DD; ++n) acc += sC[t][n] * sE[n];
      sDots[t] = acc;
    }
    float ee = 0.0f;
    if (wave == 0) {
      float pe = sP[lane] * sE[lane];
      float we = sW[lane] * sE[lane];
      float se = sS[lane] * sE[lane];
      for (int m = 16; m > 0; m >>= 1) {
        pe += __shfl_xor(pe, m);
        we += __shfl_xor(we, m);
        se += __shfl_xor(se, m);
      }
      ee = pe; wdot = we; sdot = se;
      sR[lane] = sP[lane] * (ee - sE[lane]);   // r_n = p_n*ee - p_n*e_n
    }
    __syncthreads();

    // ---- phase 2: q = (r^T T) / (1 - ee^2), zero when denom==0 ----
    if (wave == 0) {
      float num = 0.0f;
      for (int n = 0; n < DD; ++n) num += sR[n] * sT[n][lane];
      float denom = 1.0f - ee * ee;
      sQ[lane] = (denom == 0.0f) ? 0.0f : (num / denom);
    }
    __syncthreads();

    // ---- phase 3: WMMA update of ctrl rows [0,k): C <- C*T + (C e) q^T ----
    // wave w owns rows [16w, 16w+16); rows >= k are zero and stay zero.
    if (k > 0 && wave * 16 < k) {
      const int r0 = wave * 16;
      v8f acc0 = {0.f,0.f,0.f,0.f,0.f,0.f,0.f,0.f};
      v8f acc1 = {0.f,0.f,0.f,0.f,0.f,0.f,0.f,0.f};
#pragma unroll
      for (int kc = 0; kc < 8; ++kc) {
        const int k0 = kc * 4;
        v2f a;
        // A 16x4 layout: row = lane%16, k = vgpr + 2*(lane/16)
        a.x = sC[r0 + lo][k0 + 2 * hi];
        a.y = sC[r0 + lo][k0 + 2 * hi + 1];
        acc0 = __builtin_amdgcn_wmma_f32_16x16x4_f32(false, a, false, bfr[kc][0], (short)0, acc0, false, false);
        acc1 = __builtin_amdgcn_wmma_f32_16x16x4_f32(false, a, false, bfr[kc][1], (short)0, acc1, false, false);
      }
      // D 16x16 layout: row = vgpr + 8*(lane/16), col = lane%16; add rank-1 term
#pragma unroll
      for (int r = 0; r < 8; ++r) {
        int row = r0 + r + 8 * hi;
        float dq = sDots[row];
        sC[row][lo]      = acc0[r] + dq * sQ[lo];
        sC[row][lo + 16] = acc1[r] + dq * sQ[lo + 16];
      }
    }
    __syncthreads();

    // ---- phase 4: append row k; advance w, s; emit est row k (= est[:,k+1,:]) ----
    if (t < DD) sC[k][t] = sQ[t];
    if (wave == 0) {
      float wn = 0.0f, sn = 0.0f;
      for (int n = 0; n < DD; ++n) {
        wn += sW[n] * sT[n][lane];
        sn += sS[n] * sT[n][lane];
      }
      float q = sQ[lane];
      wn += wdot * q;
      sn += sdot * q + q;
      float term = fmaxf(wn - sn, 0.0f);
      float tot = term;
      for (int m = 16; m > 0; m >>= 1) tot += __shfl_xor(tot, m);
      float p = term / ((tot == 0.0f) ? 1.0f : tot);
      sW[lane] = wn; sS[lane] = sn; sP[lane] = p;
      out_est[(b * HH + k) * DD + lane] = p;
    }
    __syncthreads();

    // ---- phase 5: stream ctrls slice [k, b, :, :] as b128 stores ----
    // (rows > k are zeros in sC; slice base is 3072-float aligned)
    float4* dst4 = (float4*)(out_ctrl + (((size_t)k * BB + b) * (HH * DD)));
    for (int i = t * 4; i < HH * DD; i += NT * 4) {
      const int j = i >> 5;
      const int d = i & 31;
      float4 v;
      v.x = sC[j][d];
      v.y = sC[j][d + 1];
      v.z = sC[j][d + 2];
      v.w = sC[j][d + 3];
      dst4[i >> 2] = v;
    }
    __syncthreads();   // streaming reads must finish before next iter rewrites C
  }
}

extern "C" void kernel_launch(void* const* d_in, const int* in_sizes, int n_in,
                              void* d_out, int out_size, void* d_ws, size_t ws_size,
                              hipStream_t stream) {
  const float* lst = (const float*)d_in[0];   // (B,H,D) float32
  const float* il  = (const float*)d_in[1];   // (D,)    float32
  const float* tl  = (const float*)d_in[2];   // (D,D)   float32
  float* out_est  = (float*)d_out;                              // (B,H,D)
  float* out_ctrl = out_est + (size_t)BB * HH * DD;             // (H,B,H,D)
  dual_transition_kernel<<<dim3(BB), dim3(NT), 0, stream>>>(lst, il, tl, out_est, out_ctrl);
}